// Model_55963423867288
// MI455X (gfx1250) — compile-verified
//
#include <hip/hip_runtime.h>
#include <hip/hip_bf16.h>

typedef __attribute__((ext_vector_type(2))) float v2f;
typedef __attribute__((ext_vector_type(8))) float v8f;

// D = A(16x4,f32) * B(4x16,f32) + C(16x16,f32)  -> v_wmma_f32_16x16x4_f32
__device__ __forceinline__ v8f wmma_k4(v2f a, v2f b, v8f c) {
  return __builtin_amdgcn_wmma_f32_16x16x4_f32(
      /*neg_a=*/false, a, /*neg_b=*/false, b,
      /*c_mod=*/(short)0, c, /*reuse_a=*/false, /*reuse_b=*/false);
}

// CDNA5 async global->LDS copy (ASYNCcnt-tracked, no VGPR data path).
// GVS mode: mem = SADDR(64b) + VADDR(byte offset); LDS dest = per-lane byte offset.
__device__ __forceinline__ void async_b128(unsigned lds_off, unsigned gbyte_off,
                                           const void* base) {
  asm volatile("global_load_async_to_lds_b128 %0, %1, %2"
               :: "v"(lds_off), "v"(gbyte_off), "s"(base) : "memory");
}
__device__ __forceinline__ void wait_async0() {
  asm volatile("s_wait_asynccnt 0x0" ::: "memory");
}
__device__ __forceinline__ unsigned lds_off_of(const void* p) {
  // Generic (flat) LDS-aperture addresses carry the LDS byte offset in addr[31:0].
  return (unsigned)(unsigned long long)p;
}

// ---------------------------------------------------------------------------
// Problem dimensions (fixed by the reference)
// ---------------------------------------------------------------------------
#define BB   128   // batch
#define TT   64    // T
#define HH   1024  // H
#define HU   512   // Hu
#define JJ   129   // 2*MJW+1
#define NN   128   // 2*T
#define II   64    // I
#define MJW  64

// ---------------------------------------------------------------------------
// Kernel 1a: ts = tanh(Y @ W1 + b1)   Y:(8192,1024) W1:(1024,512) -> (8192,512)
// 64x64 tile / WG, double-buffered async global->LDS staging, fp32 WMMA.
// LDS strides 36 / 72 floats keep every 16B async chunk 16B-aligned.
// ---------------------------------------------------------------------------
#define AST 36   // A-tile row stride (floats): 144 B, 16B-aligned rows
#define BST 72   // B-tile row stride (floats): 288 B, 16B-aligned rows

__device__ __forceinline__ void stage_gemm_tiles(
    int tid, int m0, int n0, int kk,
    const float* __restrict__ Y, const float* __restrict__ W1,
    float* Asb, float* Bsb)
{
  // A tile: 64 rows x 32 floats = 512 16B-chunks; 2 per lane.
  #pragma unroll
  for (int it = 0; it < 2; ++it) {
    const int q  = it * 256 + tid;
    const int r  = q >> 3;
    const int cc = (q & 7) * 4;
    async_b128(lds_off_of(&Asb[r * AST + cc]),
               (unsigned)(((m0 + r) * HH + kk + cc) * 4), Y);
  }
  // B tile: 32 rows x 64 floats = 512 16B-chunks; 2 per lane.
  #pragma unroll
  for (int it = 0; it < 2; ++it) {
    const int q  = it * 256 + tid;
    const int r  = q >> 4;
    const int cc = (q & 15) * 4;
    async_b128(lds_off_of(&Bsb[r * BST + cc]),
               (unsigned)(((kk + r) * HU + n0 + cc) * 4), W1);
  }
}

__global__ __launch_bounds__(256) void gemm1_tanh(
    const float* __restrict__ Y, const float* __restrict__ W1,
    const float* __restrict__ b1, float* __restrict__ ts)
{
  __shared__ float As[2][64 * AST];
  __shared__ float Bs[2][32 * BST];

  const int tid  = threadIdx.x;
  const int lane = tid & 31;
  const int wave = tid >> 5;         // 0..7
  const int lo   = lane & 15;
  const int hi   = lane >> 4;        // 0 or 1
  const int m0   = blockIdx.y * 64;
  const int n0   = blockIdx.x * 64;
  const int mt   = wave & 3;         // M sub-tile 0..3
  const int ntb  = (wave >> 2) * 2;  // N sub-tile base {0,2}

  v8f acc0 = {0.f,0.f,0.f,0.f,0.f,0.f,0.f,0.f};
  v8f acc1 = {0.f,0.f,0.f,0.f,0.f,0.f,0.f,0.f};

  stage_gemm_tiles(tid, m0, n0, 0, Y, W1, As[0], Bs[0]);
  wait_async0();
  __syncthreads();

  int cur = 0;
  for (int kk = 0; kk < HH; kk += 32) {
    if (kk + 32 < HH)
      stage_gemm_tiles(tid, m0, n0, kk + 32, Y, W1, As[cur ^ 1], Bs[cur ^ 1]);

    const float* Asb = As[cur];
    const float* Bsb = Bs[cur];
    #pragma unroll
    for (int k4 = 0; k4 < 8; ++k4) {
      const int k0 = k4 * 4 + 2 * hi;
      v2f a;   a.x  = Asb[(mt * 16 + lo) * AST + k0];
               a.y  = Asb[(mt * 16 + lo) * AST + k0 + 1];
      v2f b0;  b0.x = Bsb[k0 * BST + ntb * 16 + lo];
               b0.y = Bsb[(k0 + 1) * BST + ntb * 16 + lo];
      acc0 = wmma_k4(a, b0, acc0);
      v2f b1v; b1v.x = Bsb[k0 * BST + (ntb + 1) * 16 + lo];
               b1v.y = Bsb[(k0 + 1) * BST + (ntb + 1) * 16 + lo];
      acc1 = wmma_k4(a, b1v, acc1);
    }
    wait_async0();       // my async writes for next buffer done
    __syncthreads();     // everyone done reading cur + async visible
    cur ^= 1;
  }

  #pragma unroll
  for (int v = 0; v < 8; ++v) {
    const int m  = m0 + mt * 16 + v + 8 * hi;
    const int n1 = n0 + ntb * 16 + lo;
    const int n2 = n0 + (ntb + 1) * 16 + lo;
    ts[(size_t)m * HU + n1] = tanhf(acc0[v] + b1[n1]);
    ts[(size_t)m * HU + n2] = tanhf(acc1[v] + b1[n2]);
  }
}

// ---------------------------------------------------------------------------
// Kernel 1b: per (b,t) row — head dots, softmax(129), sigmoid, banded
// transition assembly. transition[b] is 128x128 row-major in workspace.
// ---------------------------------------------------------------------------
__global__ __launch_bounds__(256) void mlp_head_transition(
    const float* __restrict__ ts, const float* __restrict__ W2,
    const float* __restrict__ b2, const float* __restrict__ W3,
    const float* __restrict__ b3, float* __restrict__ trans)
{
  __shared__ float row[HU];
  __shared__ float lg[JJ + 1];   // [0..128] softmax logits, [129] p0 logit
  __shared__ float red[2];

  const int blk = blockIdx.x;        // b*64 + t
  const int b   = blk >> 6;
  const int t   = blk & 63;
  const int tid = threadIdx.x;

  const float* tr = ts + (size_t)blk * HU;
  for (int i = tid; i < HU; i += 256) row[i] = tr[i];
  __syncthreads();

  if (tid < JJ) {
    float acc = b2[tid];
    for (int k = 0; k < HU; ++k) acc += row[k] * W2[(size_t)k * JJ + tid];
    lg[tid] = acc;
  } else if (tid == JJ) {
    float acc = b3[0];
    for (int k = 0; k < HU; ++k) acc += row[k] * W3[k];
    lg[JJ] = acc;
  }
  __syncthreads();
  if (tid == 0) {
    float m = lg[0];
    for (int j = 1; j < JJ; ++j) m = fmaxf(m, lg[j]);
    red[0] = m;
  }
  __syncthreads();
  const float mx = red[0];
  if (tid < JJ) lg[tid] = expf(lg[tid] - mx);
  __syncthreads();
  if (tid == 0) {
    float s = 0.f;
    for (int j = 0; j < JJ; ++j) s += lg[j];
    red[1] = 1.0f / s;
  }
  __syncthreads();
  const float inv = red[1];

  if (tid < NN) {
    const int n2 = tid;
    float val;
    if (n2 < TT) {
      val = lg[MJW + n2 - t] * inv;               // index in [1,127], always valid
    } else {
      val = (n2 - TT == t) ? (1.0f / (1.0f + expf(-lg[JJ]))) : 0.0f;
    }
    float* Tb = trans + (size_t)b * NN * NN;
    Tb[(size_t)t * NN + n2]        = val;
    Tb[(size_t)(t + TT) * NN + n2] = val;
  }
}

// ---------------------------------------------------------------------------
// Kernel 2: forward alignment scan. One WG (256 thr = 8 waves) per b.
// T[b], alpha[b] LDS-resident (stride 132 -> 16B-aligned rows for async B128).
// Per step: maxj via split+tree reduction, m1 = alpha@T via 2048 fp32 WMMA
// (in-place per 16-row wave band), colsum/log-sum-exp + row-normalize with
// all-thread split reductions and wave32 shuffles.
// ---------------------------------------------------------------------------
#define AS 132  // padded LDS row stride (528 B: 16B-aligned, bank-shift 4)

__global__ __launch_bounds__(256) void forward_scan(
    const float* __restrict__ trans_g, const float* __restrict__ emission,
    const int* __restrict__ src_len, float* __restrict__ out)
{
  extern __shared__ float smem[];
  float* Tl = smem;                 // 128*132
  float* Al = Tl + NN * AS;         // 128*132
  float* el = Al + NN * AS;         // 128
  float* sc = el + NN;              // 256 reduction scratch
  float* sv = sc + 256;             // scalars

  const int b    = blockIdx.x;
  const int tid  = threadIdx.x;
  const int lane = tid & 31;
  const int wave = tid >> 5;        // 0..7
  const int lo   = lane & 15;
  const int hi   = lane >> 4;
  const int band = wave * 16;       // this wave owns alpha rows [band, band+16)
  const int nh   = tid & 127;       // column/row id for split reductions
  const int half = tid >> 7;        // 0/1: which 64-wide half this thread covers
  const int sl   = src_len[b];

  // Async-stage transition[b] into LDS: 4096 x 16B chunks, 16 per lane.
  const float* Tg = trans_g + (size_t)b * NN * NN;
  #pragma unroll
  for (int it = 0; it < 16; ++it) {
    const int q  = it * 256 + tid;
    const int r  = q >> 5;
    const int cc = (q & 31) * 4;
    async_b128(lds_off_of(&Tl[r * AS + cc]), (unsigned)((r * NN + cc) * 4), Tg);
  }
  if (tid < NN) el[tid] = emission[((size_t)b * NN + tid) * II + 0];
  wait_async0();
  __syncthreads();

  // alpha0[j][n] = emission[b, n, 0] broadcast over rows j
  for (int i = tid; i < NN * NN; i += 256)
    Al[(i >> 7) * AS + (i & 127)] = el[i & 127];

  float L = 1.0f;
  if (tid == 0 && sl == 1) out[b] = 1.0f;
  __syncthreads();

  for (int step = 1; step < II; ++step) {
    // e for this step (+ prefetch next step's column)
    if (tid < NN) {
      el[tid] = emission[((size_t)b * NN + tid) * II + step];
      if (step + 1 < II)
        __builtin_prefetch(&emission[((size_t)b * NN + tid) * II + step + 1], 0, 1);
    }
    __syncthreads();

    // ---- maxj = max_n( (T[0,:] @ alpha)[n] * e[n] ) : split dot + tree max
    {
      float p = 0.f;
      const int k0 = half * 64;
      for (int k = k0; k < k0 + 64; ++k) p += Tl[k] * Al[k * AS + nh];
      sc[tid] = p;
    }
    __syncthreads();
    if (tid < NN) sc[tid] = (sc[tid] + sc[tid + 128]) * el[tid];
    __syncthreads();
    if (tid < 64) sc[tid] = fmaxf(sc[tid], sc[tid + 64]);
    __syncthreads();
    if (tid < 32) {
      float m = fmaxf(sc[tid], sc[tid + 32]);
      #pragma unroll
      for (int o = 16; o > 0; o >>= 1) m = fmaxf(m, __shfl_down(m, o));
      if (tid == 0) sv[0] = m;
    }
    __syncthreads();
    const float maxj = sv[0];

    // ---- m1 band = alpha[band,:] @ T : 8 N-tiles x 32 K4-steps of fp32 WMMA
    v8f acc[8];
    #pragma unroll
    for (int nt = 0; nt < 8; ++nt)
      acc[nt] = (v8f){0.f,0.f,0.f,0.f,0.f,0.f,0.f,0.f};
    for (int k4 = 0; k4 < 32; ++k4) {
      const int k0 = k4 * 4 + 2 * hi;
      v2f a;
      a.x = Al[(band + lo) * AS + k0];
      a.y = Al[(band + lo) * AS + k0 + 1];
      #pragma unroll
      for (int nt = 0; nt < 8; ++nt) {
        v2f bf;
        bf.x = Tl[k0 * AS + nt * 16 + lo];
        bf.y = Tl[(k0 + 1) * AS + nt * 16 + lo];
        acc[nt] = wmma_k4(a, bf, acc[nt]);
      }
    }
    // In-place tmp = m1 * e into this wave's own row band.
    #pragma unroll
    for (int nt = 0; nt < 8; ++nt) {
      const int col = nt * 16 + lo;
      const float ev = el[col];
      #pragma unroll
      for (int v = 0; v < 8; ++v)
        Al[(band + v + 8 * hi) * AS + col] = acc[nt][v] * ev;
    }
    __syncthreads();

    // ---- state_prev[n] = colsum(m1)[n] = (sum_j tmp[j][n]) / e[n]; LSE
    {
      float p = 0.f;
      const int j0 = half * 64;
      for (int j = j0; j < j0 + 64; ++j) p += Al[j * AS + nh];
      sc[tid] = p;
    }
    __syncthreads();
    if (tid < NN) {
      const float cs = sc[tid] + sc[tid + 128];
      const float sp = cs / el[tid];
      sc[tid] = expf(logf(el[tid]) + logf(sp) - maxj);
    }
    __syncthreads();
    if (tid < 64) sc[tid] += sc[tid + 64];
    __syncthreads();
    if (tid < 32) {
      float s = sc[tid] + sc[tid + 32];
      #pragma unroll
      for (int o = 16; o > 0; o >>= 1) s += __shfl_down(s, o);
      if (tid == 0) {
        L = L + logf(s) + maxj;
        if (step == sl - 1) out[b] = L;
      }
    }
    __syncthreads();  // protect sc before rownorm reuses it

    // ---- alpha <- tmp / rowsum(tmp) : two threads per row
    {
      float p = 0.f;
      const int n0c = half * 64;
      for (int n = n0c; n < n0c + 64; ++n) p += Al[nh * AS + n];
      sc[tid] = p;
    }
    __syncthreads();
    {
      const float invr = 1.0f / (sc[nh] + sc[nh + 128]);
      const int n0c = half * 64;
      for (int n = n0c; n < n0c + 64; ++n) Al[nh * AS + n] *= invr;
    }
    __syncthreads();
  }
}

// ---------------------------------------------------------------------------
// Launch
// ---------------------------------------------------------------------------
extern "C" void kernel_launch(void* const* d_in, const int* in_sizes, int n_in,
                              void* d_out, int out_size, void* d_ws, size_t ws_size,
                              hipStream_t stream)
{
  const float* y_hidden = (const float*)d_in[0];  // (128,64,1024)
  const float* emission = (const float*)d_in[1];  // (128,128,64)
  const int*   src_len  = (const int*)  d_in[2];  // (128,)
  const float* W1 = (const float*)d_in[3];        // (1024,512)
  const float* b1 = (const float*)d_in[4];        // (512,)
  const float* W2 = (const float*)d_in[5];        // (512,129)
  const float* b2 = (const float*)d_in[6];        // (129,)
  const float* W3 = (const float*)d_in[7];        // (512,1)
  const float* b3 = (const float*)d_in[8];        // (1,)
  // d_in[9]: max_jump_width == 64 (hardcoded)

  float* out = (float*)d_out;                     // (128,)
  char*  ws  = (char*)d_ws;
  float* ts    = (float*)ws;                                          // 16 MB
  float* trans = (float*)(ws + (size_t)BB * TT * HU * sizeof(float)); // 8 MB

  gemm1_tanh<<<dim3(HU / 64, (BB * TT) / 64), 256, 0, stream>>>(y_hidden, W1, b1, ts);
  mlp_head_transition<<<BB * TT, 256, 0, stream>>>(ts, W2, b2, W3, b3, trans);

  const size_t lds2 = (size_t)(2 * NN * AS + NN + 256 + 16) * sizeof(float); // ~134 KB
  (void)hipFuncSetAttribute((const void*)forward_scan,
                            hipFuncAttributeMaxDynamicSharedMemorySize, (int)lds2);
  forward_scan<<<BB, 256, lds2, stream>>>(trans, emission, src_len, out);
}